// Muti_GAT_28724741276298
// MI455X (gfx1250) — compile-verified
//
#include <hip/hip_runtime.h>

// ---------------- problem constants ----------------
#define NB 8       // batch
#define NN 256     // tokens
#define ND 512     // hidden dim
#define NE 64      // edge dim
#define NEH 64     // edge hidden dim
#define NH 4       // heads
#define NTYP 50    // dependency types
#define NEGC (-9.0e15f)

typedef __attribute__((ext_vector_type(2))) float v2f;
typedef __attribute__((ext_vector_type(8))) float v8f;

// ---------------- workspace layout (floats) ----------------
// hv      : [H,B,N,D]  = 4194304
// h_edge  : [H,B,N,D]  = 4194304
// lut     : [H,NTYP]   = 200
// asp_h   : [H,B,D]    = 16384
// asp_x   : [B,D]      = 4096
// asp_ht  : [H,B,D]    = 16384
// wgate   : [H,B]      = 32
#define WS_HV     ((size_t)0)
#define WS_HEDGE  ((size_t)4194304)
#define WS_LUT    ((size_t)8388608)
#define WS_ASPH   ((size_t)8388808)
#define WS_ASPX   ((size_t)8405192)
#define WS_ASPHT  ((size_t)8409288)
#define WS_WGATE  ((size_t)8425672)

// =====================================================================
// Wave-level f32 WMMA tile: one wave computes a 16(M) x 64(N) strip
// using V_WMMA_F32_16X16X4_F32 (exact fp32). ISA VGPR layouts:
//   A 16x4 : lane l holds A[l&15][2*(l>>4)+v]  (v2f, contiguous -> 8B load)
//   B 4x16 : lane l holds B[k + v + 2*(l>>4)][l&15]
//   C/D    : lane l, vgpr r -> row = r + 8*(l>>4), col = l&15
// =====================================================================
template <bool RELU>
__device__ __forceinline__ void gemm_wave_tile16x64(
    const float* __restrict__ A, int lda,
    const float* __restrict__ Bm, int ldb,
    float* __restrict__ Dm, int ldd,
    int K, int lane, const float* __restrict__ biasCol,
    int row0, int col0)
{
    const int m  = lane & 15;          // A row within tile / D col lane
    const int kh = (lane >> 4) << 1;   // 0 or 2: K sub-offset for this half-wave
    const int n  = lane & 15;

    v8f acc[4];
#pragma unroll
    for (int q = 0; q < 4; ++q) {
        float bv = biasCol ? biasCol[col0 + 16 * q + n] : 0.0f;
#pragma unroll
        for (int r = 0; r < 8; ++r) acc[q][r] = bv;
    }

    const float* arow = A + (size_t)(row0 + m) * lda + kh;

    for (int k0 = 0; k0 < K; k0 += 4) {
        v2f a = *reinterpret_cast<const v2f*>(arow + k0);       // 8B aligned
        const float* b0 = Bm + (size_t)(k0 + kh) * ldb + col0 + n;
        const float* b1 = b0 + ldb;
#pragma unroll
        for (int q = 0; q < 4; ++q) {
            v2f bfrag;
            bfrag[0] = b0[16 * q];
            bfrag[1] = b1[16 * q];
            acc[q] = __builtin_amdgcn_wmma_f32_16x16x4_f32(
                false, a, false, bfrag, (short)0, acc[q], false, false);
        }
    }

    const int mg = (lane >> 4) * 8;
#pragma unroll
    for (int q = 0; q < 4; ++q) {
#pragma unroll
        for (int r = 0; r < 8; ++r) {
            float v = acc[q][r];
            if (RELU) v = fmaxf(v, 0.0f);
            Dm[(size_t)(row0 + mg + r) * ldd + col0 + 16 * q + n] = v;
        }
    }
}

// =====================================================================
// Kernel 1: edge-score LUT  lut[h][t] = Wl_h . relu(edge_table[t] @ Wm_h)
// (t==0 masked to 0 by the dependency mask)
// =====================================================================
__global__ void k_lut(const float* __restrict__ table,
                      const float* __restrict__ Wm,
                      const float* __restrict__ Wl,
                      float* __restrict__ lut)
{
    int id = blockIdx.x * blockDim.x + threadIdx.x;
    if (id >= NH * NTYP) return;
    int h = id / NTYP, t = id % NTYP;
    float acc = 0.0f;
    if (t != 0) {
        const float* et = table + (size_t)t * NE;
        const float* wm = Wm + (size_t)h * NE * NEH;
        const float* wl = Wl + (size_t)h * NEH;
        for (int e2 = 0; e2 < NEH; ++e2) {
            float s = 0.0f;
            for (int e = 0; e < NE; ++e) s += et[e] * wm[e * NEH + e2];
            acc += wl[e2] * fmaxf(s, 0.0f);
        }
    }
    lut[id] = acc;
}

// =====================================================================
// Kernel 2: per-row attention. One block = one (h,b,i) row of length N.
// ew -> mean -> denom -> mask -> softmax.  att written straight to d_out.
// =====================================================================
__global__ void k_att(const int* __restrict__ adj,
                      const int* __restrict__ dep,
                      const float* __restrict__ lut,
                      float* __restrict__ att)
{
    __shared__ float lut_s[64];
    __shared__ float red[NN];
    const int j   = threadIdx.x;
    const int blk = blockIdx.x;
    const int h   = blk / (NB * NN);
    const int rem = blk % (NB * NN);
    const int b   = rem / NN;
    const int i   = rem % NN;

    if (j < NTYP) lut_s[j] = lut[h * NTYP + j];
    __syncthreads();

    const size_t rbase = ((size_t)b * NN + i) * NN;
    const int t  = dep[rbase + j];
    const float ew = lut_s[t];

    // mean over j
    red[j] = ew;
    __syncthreads();
    for (int s = NN / 2; s > 0; s >>= 1) {
        if (j < s) red[j] += red[j + s];
        __syncthreads();
    }
    const float denom = fabsf(red[0] / (float)NN) + 1e-10f;
    __syncthreads();

    const float score = (adj[rbase + j] != 0) ? (ew / denom) : NEGC;

    // softmax: max
    red[j] = score;
    __syncthreads();
    for (int s = NN / 2; s > 0; s >>= 1) {
        if (j < s) red[j] = fmaxf(red[j], red[j + s]);
        __syncthreads();
    }
    const float mx = red[0];
    __syncthreads();

    const float e = expf(score - mx);
    red[j] = e;
    __syncthreads();
    for (int s = NN / 2; s > 0; s >>= 1) {
        if (j < s) red[j] += red[j + s];
        __syncthreads();
    }
    const float sum = red[0];

    att[(((size_t)h * NB + b) * NN + i) * NN + j] = e / sum;
}

// =====================================================================
// Kernel 3: hv[h] = x @ Wto[h] + bto[h]   (WMMA, M=2048 N=512 K=512)
// 8 waves/block, wave -> 16x64 strip. 4096 wave-tiles total.
// =====================================================================
__global__ void k_gemm_hv(const float* __restrict__ x,
                          const float* __restrict__ Wto,
                          const float* __restrict__ bto,
                          float* __restrict__ hv)
{
    const int wv   = blockIdx.x * (blockDim.x >> 5) + (threadIdx.x >> 5);
    const int lane = threadIdx.x & 31;
    const int h    = wv >> 10;           // 1024 wave-tiles per head
    const int rem  = wv & 1023;
    const int mt   = rem >> 3;           // 128 M-tiles of 16
    const int nt   = rem & 7;            // 8 N-tiles of 64
    gemm_wave_tile16x64<false>(
        x, ND,
        Wto + (size_t)h * ND * ND, ND,
        hv + (size_t)h * NB * NN * ND, ND,
        ND, lane, bto + (size_t)h * ND,
        mt * 16, nt * 64);
}

// =====================================================================
// Kernel 4: h_edge[h,b] = relu(att[h,b] @ hv[h,b])  (WMMA, M=256 N=512 K=256)
// =====================================================================
__global__ void k_gemm_hedge(const float* __restrict__ att,
                             const float* __restrict__ hv,
                             float* __restrict__ h_edge)
{
    const int wv   = blockIdx.x * (blockDim.x >> 5) + (threadIdx.x >> 5);
    const int lane = threadIdx.x & 31;
    const int h    = wv >> 10;           // 1024 wave-tiles per head (8 b * 128)
    const int rem  = wv & 1023;
    const int b    = rem >> 7;           // 128 tiles per (h,b)
    const int t    = rem & 127;
    const int mt   = t >> 3;             // 16 M-tiles
    const int nt   = t & 7;              // 8 N-tiles
    const size_t hb = (size_t)h * NB + b;
    gemm_wave_tile16x64<true>(
        att + hb * NN * NN, NN,
        hv + hb * NN * ND, ND,
        h_edge + hb * NN * ND, ND,
        NN, lane, nullptr,
        mt * 16, nt * 64);
}

// =====================================================================
// Kernel 5a/5b: span max-pools
// =====================================================================
__global__ void k_aspect_h(const float* __restrict__ h_edge,
                           const int* __restrict__ aidx,
                           float* __restrict__ asp_h)
{
    int id = blockIdx.x * blockDim.x + threadIdx.x;   // H*B*D
    if (id >= NH * NB * ND) return;
    int d  = id % ND;
    int hb = id / ND;
    int b  = hb % NB;
    const int lo = aidx[b * 2 + 0], hi = aidx[b * 2 + 1];
    float m = 0.0f;   // out-of-span entries contribute exactly 0 (product form)
    const float* base = h_edge + (size_t)hb * NN * ND + d;
    for (int n = 0; n < NN; ++n) {
        float v = (n >= lo && n <= hi) ? base[(size_t)n * ND] : 0.0f;
        m = fmaxf(m, v);
    }
    asp_h[id] = m;
}

__global__ void k_aspect_x(const float* __restrict__ x,
                           const int* __restrict__ aidx,
                           float* __restrict__ asp_x)
{
    int id = blockIdx.x * blockDim.x + threadIdx.x;   // B*D
    if (id >= NB * ND) return;
    int d = id % ND;
    int b = id / ND;
    const int lo = aidx[b * 2 + 0], hi = aidx[b * 2 + 1];
    float m = -3.4e38f;
    const float* base = x + (size_t)b * NN * ND + d;
    for (int n = 0; n < NN; ++n) {
        float v = (n >= lo && n <= hi) ? base[(size_t)n * ND] : 0.0f;
        m = fmaxf(m, v);
    }
    asp_x[id] = m;
}

// =====================================================================
// Kernel 6: asp_ht = tanh(asp_h @ Wasp + basp)   (tiny: 32x512x512)
// =====================================================================
__global__ void k_aspect_lin(const float* __restrict__ asp_h,
                             const float* __restrict__ Wasp,
                             const float* __restrict__ basp,
                             float* __restrict__ asp_ht)
{
    int id = blockIdx.x * blockDim.x + threadIdx.x;   // H*B*D
    if (id >= NH * NB * ND) return;
    int dp = id % ND;
    int hb = id / ND;
    float acc = basp[dp];
    const float* row = asp_h + (size_t)hb * ND;
    for (int d = 0; d < ND; ++d) acc += row[d] * Wasp[(size_t)d * ND + dp];
    asp_ht[id] = tanhf(acc);
}

// =====================================================================
// Kernel 7: head gate w[h,b] = relu([asp_x ; asp_ht] . Ww + bw)
// one block per (h,b)
// =====================================================================
__global__ void k_headw(const float* __restrict__ asp_x,
                        const float* __restrict__ asp_ht,
                        const float* __restrict__ Ww,
                        const float* __restrict__ bw,
                        float* __restrict__ wgate)
{
    __shared__ float red[256];
    const int hb = blockIdx.x;            // H*B
    const int b  = hb % NB;
    float p = 0.0f;
    for (int d = threadIdx.x; d < ND; d += blockDim.x)
        p += asp_x[(size_t)b * ND + d] * Ww[d] +
             asp_ht[(size_t)hb * ND + d] * Ww[ND + d];
    red[threadIdx.x] = p;
    __syncthreads();
    for (int s = 128; s > 0; s >>= 1) {
        if (threadIdx.x < s) red[threadIdx.x] += red[threadIdx.x + s];
        __syncthreads();
    }
    if (threadIdx.x == 0) wgate[hb] = fmaxf(red[0] + bw[0], 0.0f);
}

// Kernel 8: softmax over heads per batch (32 values)
__global__ void k_headsm(float* __restrict__ wgate)
{
    int b = threadIdx.x;
    if (b >= NB) return;
    float v[NH], mx = -3.4e38f;
    for (int h = 0; h < NH; ++h) { v[h] = wgate[h * NB + b]; mx = fmaxf(mx, v[h]); }
    float s = 0.0f;
    for (int h = 0; h < NH; ++h) { v[h] = expf(v[h] - mx); s += v[h]; }
    for (int h = 0; h < NH; ++h) wgate[h * NB + b] = v[h] / s;
}

// =====================================================================
// Kernel 9: h_prime = relu(mean_h wgate[h,b] * h_edge[h,b,n,d])
// =====================================================================
__global__ void k_final(const float* __restrict__ h_edge,
                        const float* __restrict__ wgate,
                        float* __restrict__ out)
{
    size_t id = (size_t)blockIdx.x * blockDim.x + threadIdx.x;   // B*N*D
    if (id >= (size_t)NB * NN * ND) return;
    int b = (int)(id / ((size_t)NN * ND));
    size_t nd = id % ((size_t)NN * ND);
    float acc = 0.0f;
#pragma unroll
    for (int h = 0; h < NH; ++h)
        acc += wgate[h * NB + b] * h_edge[(((size_t)h * NB + b) * NN * ND) + nd];
    out[id] = fmaxf(acc * (1.0f / NH), 0.0f);
}

// =====================================================================
extern "C" void kernel_launch(void* const* d_in, const int* in_sizes, int n_in,
                              void* d_out, int out_size, void* d_ws, size_t ws_size,
                              hipStream_t stream)
{
    const float* x      = (const float*)d_in[0];
    const int*   adj    = (const int*)  d_in[1];
    const int*   dep    = (const int*)  d_in[2];
    const int*   aidx   = (const int*)  d_in[3];
    const float* table  = (const float*)d_in[4];
    const float* Wm     = (const float*)d_in[5];
    const float* Wl     = (const float*)d_in[6];
    const float* Wto    = (const float*)d_in[7];
    const float* bto    = (const float*)d_in[8];
    const float* Wasp   = (const float*)d_in[9];
    const float* basp   = (const float*)d_in[10];
    const float* Ww     = (const float*)d_in[11];
    const float* bw     = (const float*)d_in[12];

    float* out     = (float*)d_out;
    float* out_hp  = out;                               // [B,N,D]
    float* out_att = out + (size_t)NB * NN * ND;        // [H,B,N,N]

    float* ws      = (float*)d_ws;
    float* hv      = ws + WS_HV;
    float* h_edge  = ws + WS_HEDGE;
    float* lut     = ws + WS_LUT;
    float* asp_h   = ws + WS_ASPH;
    float* asp_x   = ws + WS_ASPX;
    float* asp_ht  = ws + WS_ASPHT;
    float* wgate   = ws + WS_WGATE;

    // 1. edge-score LUT (collapses the 17-GFLOP edge MLP to 50 entries/head)
    k_lut<<<1, 256, 0, stream>>>(table, Wm, Wl, lut);

    // 2. attention rows -> att written directly into d_out's att slice
    k_att<<<NH * NB * NN, NN, 0, stream>>>(adj, dep, lut, out_att);

    // 3. hv = x @ Wto + bto   (f32 WMMA)
    k_gemm_hv<<<512, 256, 0, stream>>>(x, Wto, bto, hv);

    // 4. h_edge = relu(att @ hv)  (f32 WMMA)
    k_gemm_hedge<<<512, 256, 0, stream>>>(out_att, hv, h_edge);

    // 5. span max-pools
    k_aspect_h<<<(NH * NB * ND + 255) / 256, 256, 0, stream>>>(h_edge, aidx, asp_h);
    k_aspect_x<<<(NB * ND + 255) / 256, 256, 0, stream>>>(x, aidx, asp_x);

    // 6. tanh(asp_h @ Wasp + basp)
    k_aspect_lin<<<(NH * NB * ND + 255) / 256, 256, 0, stream>>>(asp_h, Wasp, basp, asp_ht);

    // 7-8. head gate + softmax over heads
    k_headw<<<NH * NB, 256, 0, stream>>>(asp_x, asp_ht, Ww, bw, wgate);
    k_headsm<<<1, 32, 0, stream>>>(wgate);

    // 9. weighted mean over heads + relu -> h_prime
    k_final<<<(NB * NN * ND + 255) / 256, 256, 0, stream>>>(h_edge, wgate, out_hp);
}